// ScoreFPELoss_45749991637553
// MI455X (gfx1250) — compile-verified
//
#include <hip/hip_runtime.h>
#include <math.h>

// Problem constants (B=1024, D=64, H=512); K for layer 1 padded 65 -> 80 (mult of BK=16).
#define B_ 1024
#define D_ 64
#define H_ 512
#define KP 80

// GEMM tiling
#define BM 64
#define BN 64
#define BK 16
#define ASTRIDE 20   // padded LDS row stride for A tile (BK + 4): conflict-free A reads
#define BSTRIDE 80   // padded LDS row stride for B tile (BN + 16): disjoint bank halves
#define AOFF 0
#define BOFF (BM * ASTRIDE)
#define STAGE_F (BM * ASTRIDE + BK * BSTRIDE)  // floats per stage

typedef float v2f __attribute__((ext_vector_type(2)));
typedef float v8f __attribute__((ext_vector_type(8)));

// ----------------------------------------------------------------------------
// fp32 WMMA GEMM with async global->LDS double-buffered staging.
// Block: 256 threads (8 waves) -> 64x64 output tile; wave (wm,wn) owns 16x32
// (two 16x16 WMMA accumulators). K-loop: BK=16 per stage, 4 x v_wmma_f32_16x16x4_f32
// per accumulator per stage.
// WMMA VGPR layouts per CDNA5 ISA 7.12.2:
//   A 16x4 : lane<16 -> M=lane, K={k,k+1}; lane>=16 -> M=lane-16, K={k+2,k+3}
//   B 4x16 : v0 = row k (lanes 0-15) / row k+2 (lanes 16-31); v1 = k+1 / k+3
//   C/D    : vgpr r -> M = r + 8*(lane>=16), N = lane&15
// ----------------------------------------------------------------------------
template <bool BIAS, bool TANH>
__global__ __launch_bounds__(256) void gemm_wmma_lds(
    const float* __restrict__ A, const float* __restrict__ Bm,
    const float* __restrict__ bias,
    float* __restrict__ Y, float* __restrict__ Dout,
    int M, int N, int K) {
  __shared__ float lds[2][STAGE_F];

  const int tid  = threadIdx.x;
  const int lane = tid & 31;
  const int wid  = tid >> 5;           // 0..7
  const int tilesN = N / BN;
  const int bm = (int)(blockIdx.x / tilesN) * BM;
  const int bn = (int)(blockIdx.x % tilesN) * BN;

  // per-thread async b128 load assignment (one for A, one for B per stage)
  const int ar = tid >> 2;             // 0..63  A-tile row
  const int ak = (tid & 3) << 2;       // 0,4,8,12  A-tile col segment
  const int br = tid >> 4;             // 0..15  B-tile row
  const int bc = (tid & 15) << 2;      // 0..60  B-tile col segment

  const int wm   = (wid & 3) * 16;     // wave M offset inside tile
  const int wn   = (wid >> 2) * 32;    // wave N offset inside tile (0 or 32)
  const int half = lane >> 4;
  const int l16  = lane & 15;

  const float* gA = A + (size_t)(bm + ar) * K + ak;
  const float* gB = Bm + (size_t)br * N + bn + bc;

  // LDS byte addresses for the async loads (low 32 bits of generic LDS ptr)
  uint32_t ldsA[2], ldsB[2];
#pragma unroll
  for (int s = 0; s < 2; ++s) {
    ldsA[s] = (uint32_t)(uintptr_t)&lds[s][AOFF + ar * ASTRIDE + ak];
    ldsB[s] = (uint32_t)(uintptr_t)&lds[s][BOFF + br * BSTRIDE + bc];
  }

  auto stage_load = [&](int s, int k0) {
    const float* pa = gA + k0;                 // A[bm+ar][k0+ak .. +3]
    const float* pb = gB + (size_t)k0 * N;     // B[k0+br][bn+bc .. +3]
    asm volatile("global_load_async_to_lds_b128 %0, %1, off"
                 :: "v"(ldsA[s]), "v"(pa) : "memory");
    asm volatile("global_load_async_to_lds_b128 %0, %1, off"
                 :: "v"(ldsB[s]), "v"(pb) : "memory");
  };

  v8f acc0 = {}, acc1 = {};
  stage_load(0, 0);

  const int nK = K / BK;
  for (int ko = 0; ko < nK; ++ko) {
    const int cur = ko & 1;
    asm volatile("s_wait_asynccnt 0" ::: "memory");  // my stage-cur loads done
    __syncthreads();                                 // everyone's loads visible
    if (ko + 1 < nK) stage_load(cur ^ 1, (ko + 1) * BK);  // overlap next stage

    const float* As = &lds[cur][AOFF];
    const float* Bs = &lds[cur][BOFF];
#pragma unroll
    for (int kk = 0; kk < BK; kk += 4) {
      const int kb = kk + half * 2;
      v2f a, b0, b1;
      a.x  = As[(wm + l16) * ASTRIDE + kb];
      a.y  = As[(wm + l16) * ASTRIDE + kb + 1];
      b0.x = Bs[kb * BSTRIDE + wn + l16];
      b0.y = Bs[(kb + 1) * BSTRIDE + wn + l16];
      b1.x = Bs[kb * BSTRIDE + wn + 16 + l16];
      b1.y = Bs[(kb + 1) * BSTRIDE + wn + 16 + l16];
      acc0 = __builtin_amdgcn_wmma_f32_16x16x4_f32(false, a, false, b0,
                                                   (short)0, acc0, false, false);
      acc1 = __builtin_amdgcn_wmma_f32_16x16x4_f32(false, a, false, b1,
                                                   (short)0, acc1, false, false);
    }
    __syncthreads();  // all waves done reading stage cur before it is refilled
  }

  const int gn0 = bn + wn + l16;
  const float bv0 = BIAS ? bias[gn0] : 0.0f;
  const float bv1 = BIAS ? bias[gn0 + 16] : 0.0f;
#pragma unroll
  for (int r = 0; r < 8; ++r) {
    const int gm = bm + wm + half * 8 + r;
    float v0 = acc0[r] + bv0;
    float v1 = acc1[r] + bv1;
    if (TANH) {
      float h0 = tanhf(v0), h1 = tanhf(v1);
      Y[(size_t)gm * N + gn0]         = h0;
      Y[(size_t)gm * N + gn0 + 16]    = h1;
      Dout[(size_t)gm * N + gn0]      = 1.0f - h0 * h0;
      Dout[(size_t)gm * N + gn0 + 16] = 1.0f - h1 * h1;
    } else {
      Y[(size_t)gm * N + gn0]      = v0;
      Y[(size_t)gm * N + gn0 + 16] = v1;
    }
  }
}

// ---------------------------- prep kernels ----------------------------------
__global__ void k_build_z(const float* __restrict__ x, const float* __restrict__ t,
                          float* __restrict__ Z) {
  int i = blockIdx.x * blockDim.x + threadIdx.x;
  if (i >= B_ * KP) return;
  int b = i / KP, c = i % KP;
  Z[i] = (c < D_) ? x[b * D_ + c] : (c == D_ ? t[b] : 0.0f);
}

__global__ void k_build_w1p(const float* __restrict__ W1, float* __restrict__ W1p) {
  int i = blockIdx.x * blockDim.x + threadIdx.x;
  if (i >= KP * H_) return;
  int r = i / H_;
  W1p[i] = (r <= D_) ? W1[i] : 0.0f;  // rows 0..64 copy, 65..79 zero pad
}

// C[k,j] = W2[k,j] * sum_i W1[i,k]*W3[j,i]; also CT = C^T.
__global__ void k_build_C(const float* __restrict__ W1, const float* __restrict__ W2,
                          const float* __restrict__ W3, float* __restrict__ C,
                          float* __restrict__ CT) {
  int i = blockIdx.x * blockDim.x + threadIdx.x;
  if (i >= H_ * H_) return;
  int k = i / H_, j = i % H_;
  float acc = 0.0f;
#pragma unroll 8
  for (int ii = 0; ii < D_; ++ii) acc += W1[ii * H_ + k] * W3[j * D_ + ii];
  float c = W2[i] * acc;
  C[i] = c;
  CT[(size_t)j * H_ + k] = c;
}

__global__ void k_transpose512(const float* __restrict__ Wsrc, float* __restrict__ Wdst) {
  int i = blockIdx.x * blockDim.x + threadIdx.x;
  if (i >= H_ * H_) return;
  int k = i / H_, j = i % H_;
  Wdst[(size_t)j * H_ + k] = Wsrc[i];
}

// W3T[ii,j]=W3[j,ii]; W1xT[k,m]=W1[m,k]; w3s[j]=sum_i W3[j,i]
__global__ void k_small_prep(const float* __restrict__ W3, const float* __restrict__ W1,
                             float* __restrict__ W3T, float* __restrict__ W1xT,
                             float* __restrict__ w3s) {
  int i = blockIdx.x * blockDim.x + threadIdx.x;
  if (i < D_ * H_) {
    int ii = i / H_, j = i % H_;
    W3T[i] = W3[j * D_ + ii];
  }
  if (i < H_ * D_) {
    int k = i / D_, m = i % D_;
    W1xT[i] = W1[m * H_ + k];
  }
  if (i < H_) {
    float a = 0.0f;
    for (int ii = 0; ii < D_; ++ii) a += W3[i * D_ + ii];
    w3s[i] = a;
  }
}

// ------------------------- elementwise kernels ------------------------------
__global__ void k_dh1(const float* __restrict__ D1, const float* __restrict__ W1p,
                      float* __restrict__ DH1) {
  int i = blockIdx.x * blockDim.x + threadIdx.x;
  if (i >= B_ * H_) return;
  DH1[i] = D1[i] * W1p[D_ * H_ + (i & (H_ - 1))];  // broadcast t-row of W1
}

__global__ void k_mul_inplace(float* __restrict__ Y, const float* __restrict__ Dm, int n) {
  int i = blockIdx.x * blockDim.x + threadIdx.x;
  if (i < n) Y[i] *= Dm[i];
}

__global__ void k_v(const float* __restrict__ S, const float* __restrict__ x,
                    float* __restrict__ V) {
  int i = blockIdx.x * blockDim.x + threadIdx.x;
  if (i < B_ * D_) V[i] = 2.0f * S[i] + x[i];
}

__global__ void k_r1t2(const float* __restrict__ H1, const float* __restrict__ D1,
                       const float* __restrict__ H2, const float* __restrict__ D2,
                       float* __restrict__ P, float* __restrict__ Q) {
  int i = blockIdx.x * blockDim.x + threadIdx.x;
  if (i >= B_ * H_) return;
  P[i] = -2.0f * H1[i] * D1[i] * P[i];  // r1
  Q[i] = -2.0f * H2[i] * D2[i] * Q[i];  // t2
}

__global__ void k_u(float* __restrict__ GA1, const float* __restrict__ R1,
                    const float* __restrict__ D1, const float* __restrict__ R2) {
  int i = blockIdx.x * blockDim.x + threadIdx.x;
  if (i >= B_ * H_) return;
  GA1[i] = GA1[i] + R1[i] + D1[i] * R2[i];  // U
}

// s_t[b] = sum_j D2*DA2*w3s  (LDS tree reduce; one 256-thread block per row)
__global__ void k_st(const float* __restrict__ D2, const float* __restrict__ DA2,
                     const float* __restrict__ w3s, float* __restrict__ st) {
  __shared__ float red[256];
  int b = blockIdx.x;
  float a = 0.0f;
  for (int j = threadIdx.x; j < H_; j += 256)
    a += D2[(size_t)b * H_ + j] * DA2[(size_t)b * H_ + j] * w3s[j];
  red[threadIdx.x] = a;
  __syncthreads();
  for (int sft = 128; sft > 0; sft >>= 1) {
    if ((int)threadIdx.x < sft) red[threadIdx.x] += red[threadIdx.x + sft];
    __syncthreads();
  }
  if (threadIdx.x == 0) st[b] = red[0];
}

// loss[b] = mean_i |s_t[b] - 0.5*beta[b]*(GX[b,i]+S[b,i])|
__global__ void k_loss(const float* __restrict__ st, const float* __restrict__ beta,
                       const float* __restrict__ GX, const float* __restrict__ S,
                       float* __restrict__ out) {
  __shared__ float red[64];
  int b = blockIdx.x, i = threadIdx.x;
  float g = GX[b * D_ + i] + S[b * D_ + i];
  red[i] = fabsf(st[b] - 0.5f * beta[b] * g);
  __syncthreads();
  for (int sft = 32; sft > 0; sft >>= 1) {
    if (i < sft) red[i] += red[i + sft];
    __syncthreads();
  }
  if (i == 0) out[b] = red[0] * (1.0f / 64.0f);
}

// ------------------------------- launcher -----------------------------------
static inline void launch_gemm(const float* A, const float* Bm, const float* bias,
                               float* Y, float* Dout, int M, int N, int K,
                               bool biasF, bool tanhF, hipStream_t s) {
  int blocks = (M / BM) * (N / BN);
  if (tanhF)
    gemm_wmma_lds<true, true><<<blocks, 256, 0, s>>>(A, Bm, bias, Y, Dout, M, N, K);
  else if (biasF)
    gemm_wmma_lds<true, false><<<blocks, 256, 0, s>>>(A, Bm, bias, Y, nullptr, M, N, K);
  else
    gemm_wmma_lds<false, false><<<blocks, 256, 0, s>>>(A, Bm, nullptr, Y, nullptr, M, N, K);
}

extern "C" void kernel_launch(void* const* d_in, const int* in_sizes, int n_in,
                              void* d_out, int out_size, void* d_ws, size_t ws_size,
                              hipStream_t stream) {
  const float* x    = (const float*)d_in[0];
  const float* t    = (const float*)d_in[1];
  const float* beta = (const float*)d_in[2];
  const float* W1   = (const float*)d_in[3];
  const float* b1   = (const float*)d_in[4];
  const float* W2   = (const float*)d_in[5];
  const float* b2   = (const float*)d_in[6];
  const float* W3   = (const float*)d_in[7];
  const float* b3   = (const float*)d_in[8];
  float* out = (float*)d_out;
  float* w   = (float*)d_ws;

  // workspace carve-up (floats); total ~27.8 MB
  size_t o = 0;
  float* Z    = w + o; o += (size_t)B_ * KP;
  float* W1p  = w + o; o += (size_t)KP * H_;
  float* C    = w + o; o += (size_t)H_ * H_;
  float* CT   = w + o; o += (size_t)H_ * H_;
  float* W2T  = w + o; o += (size_t)H_ * H_;
  float* W3T  = w + o; o += (size_t)D_ * H_;
  float* W1xT = w + o; o += (size_t)H_ * D_;
  float* w3s  = w + o; o += (size_t)H_;
  float* H1   = w + o; o += (size_t)B_ * H_;
  float* D1v  = w + o; o += (size_t)B_ * H_;
  float* H2   = w + o; o += (size_t)B_ * H_;
  float* D2v  = w + o; o += (size_t)B_ * H_;
  float* DH1  = w + o; o += (size_t)B_ * H_;
  float* DA2  = w + o; o += (size_t)B_ * H_;
  float* GH2  = w + o; o += (size_t)B_ * H_;   // -> GA2 in place
  float* GH1  = w + o; o += (size_t)B_ * H_;   // -> GA1 -> U in place
  float* Pb   = w + o; o += (size_t)B_ * H_;   // -> R1 in place
  float* Qb   = w + o; o += (size_t)B_ * H_;   // -> T2 in place
  float* R2   = w + o; o += (size_t)B_ * H_;
  float* Sb   = w + o; o += (size_t)B_ * D_;
  float* Vb   = w + o; o += (size_t)B_ * D_;
  float* GX   = w + o; o += (size_t)B_ * D_;
  float* st   = w + o; o += (size_t)B_;

  const int EW = 256;
  auto nb = [](int n) { return (n + 255) / 256; };

  // ---- prep (sample independent) ----
  k_build_z<<<nb(B_ * KP), EW, 0, stream>>>(x, t, Z);
  k_build_w1p<<<nb(KP * H_), EW, 0, stream>>>(W1, W1p);
  k_build_C<<<nb(H_ * H_), EW, 0, stream>>>(W1, W2, W3, C, CT);
  k_transpose512<<<nb(H_ * H_), EW, 0, stream>>>(W2, W2T);
  k_small_prep<<<nb(D_ * H_), EW, 0, stream>>>(W3, W1, W3T, W1xT, w3s);

  // ---- forward ----
  launch_gemm(Z, W1p, b1, H1, D1v, B_, H_, KP, true, true, stream);      // h1,d1
  launch_gemm(H1, W2, b2, H2, D2v, B_, H_, H_, true, true, stream);      // h2,d2
  launch_gemm(H2, W3, b3, Sb, nullptr, B_, D_, H_, true, false, stream); // s

  // ---- s_t = (d2 ⊙ (d1⊙W1[t-row])W2) · rowsum(W3) ----
  k_dh1<<<nb(B_ * H_), EW, 0, stream>>>(D1v, W1p, DH1);
  launch_gemm(DH1, W2, nullptr, DA2, nullptr, B_, H_, H_, false, false, stream);
  k_st<<<B_, 256, 0, stream>>>(D2v, DA2, w3s, st);

  // ---- VJP: gx_vjp = W1x (d1 ⊙ W2 (d2 ⊙ W3 (2s+x))) ----
  k_v<<<nb(B_ * D_), EW, 0, stream>>>(Sb, x, Vb);
  launch_gemm(Vb, W3T, nullptr, GH2, nullptr, B_, H_, D_, false, false, stream);
  k_mul_inplace<<<nb(B_ * H_), EW, 0, stream>>>(GH2, D2v, B_ * H_);      // GA2
  launch_gemm(GH2, W2T, nullptr, GH1, nullptr, B_, H_, H_, false, false, stream);
  k_mul_inplace<<<nb(B_ * H_), EW, 0, stream>>>(GH1, D1v, B_ * H_);      // GA1

  // ---- grad of div = d1ᵀ C d2 ----
  launch_gemm(D2v, CT, nullptr, Pb, nullptr, B_, H_, H_, false, false, stream); // p = C d2
  launch_gemm(D1v, C, nullptr, Qb, nullptr, B_, H_, H_, false, false, stream);  // q = Cᵀ d1
  k_r1t2<<<nb(B_ * H_), EW, 0, stream>>>(H1, D1v, H2, D2v, Pb, Qb);      // r1, t2
  launch_gemm(Qb, W2T, nullptr, R2, nullptr, B_, H_, H_, false, false, stream); // r2' = W2 t2
  k_u<<<nb(B_ * H_), EW, 0, stream>>>(GH1, Pb, D1v, R2);                 // U

  // ---- gx = U @ W1xᵀ ; g = gx + s ; loss ----
  launch_gemm(GH1, W1xT, nullptr, GX, nullptr, B_, D_, H_, false, false, stream);
  k_loss<<<B_, 64, 0, stream>>>(st, beta, GX, Sb, out);
}